// LogisticModel_12154757448340
// MI455X (gfx1250) — compile-verified
//
#include <hip/hip_runtime.h>

typedef float v2f __attribute__((ext_vector_type(2)));
typedef float v4f __attribute__((ext_vector_type(4)));
typedef float v8f __attribute__((ext_vector_type(8)));

#define DECAY 0.9f
// per-element additive constant: -log(0.1) - 0.5*log(2*pi)
#define LOGP_CONST 1.3836465597893728f

__device__ __forceinline__ float sigmoid_fast(float v) {
  // sigma(v) = 1 / (1 + e^-v); hardware v_exp_f32 + v_rcp_f32
  return __builtin_amdgcn_rcpf(1.0f + __expf(-v));
}

__global__ __launch_bounds__(256) void logistic_logp_kernel(
    const float* __restrict__ s, const float* __restrict__ x,
    float* __restrict__ out, int T) {
  const int row  = blockIdx.x;
  const int tid  = threadIdx.x;
  const int lane = tid & 31;
  const long long rowoff = (long long)row * (long long)T;
  const float* __restrict__ sr = s + rowoff;
  const float* __restrict__ xr = x + rowoff;
  const int TV = T >> 2;  // float4 vectors per row

  float acc = 0.0f;

  for (int v = tid; v < TV; v += 256) {
    // streaming (non-temporal) b128 loads: data is touched exactly once
    v4f xv = __builtin_nontemporal_load((const v4f*)xr + v);
    v4f sv = __builtin_nontemporal_load((const v4f*)sr + v);

    // prefetch next tile (speculative; OOB prefetches are dropped)
    __builtin_prefetch(xr + 4 * (v + 256), 0, 1);
    __builtin_prefetch(sr + 4 * (v + 256), 0, 1);

    // x[4v-1]: neighbor lane's last element; lane 0 falls back to one
    // scalar load (or 0 at the t==0 boundary).
    float prevw = __shfl_up(xv.w, 1, 32);
    float xm1;
    if (lane == 0) {
      xm1 = (v == 0) ? 0.0f : xr[4 * v - 1];
    } else {
      xm1 = prevw;
    }

    float r0 = xv.x - DECAY * xm1  - sigmoid_fast(sv.x);
    float r1 = xv.y - DECAY * xv.x - sigmoid_fast(sv.y);
    float r2 = xv.z - DECAY * xv.y - sigmoid_fast(sv.z);
    float r3 = xv.w - DECAY * xv.z - sigmoid_fast(sv.w);

    // -0.5*(r/0.1)^2 == -50*r^2
    acc = fmaf(-50.0f * r0, r0, acc);
    acc = fmaf(-50.0f * r1, r1, acc);
    acc = fmaf(-50.0f * r2, r2, acc);
    acc = fmaf(-50.0f * r3, r3, acc);
  }

  // ---- block reduction of 256 partials via chained V_WMMA_F32_16X16X4_F32 ----
  __shared__ float parts[256];
  parts[tid] = acc;
  __syncthreads();

  if (tid < 32) {  // wave 0: fully active -> EXEC all ones (WMMA requirement)
    v2f bones;
    bones.x = 1.0f;  // B = ones(4x16): every B VGPR slot is 1.0
    bones.y = 1.0f;
    v8f c = {};
#pragma unroll
    for (int i = 0; i < 4; ++i) {
      // A-matrix 16x4 f32 layout: lanes 0-15 hold K=0,1 in v0,v1;
      // lanes 16-31 hold K=2,3.  A[m,k] = parts[64*i + 4*m + k].
      const int base = i * 64;
      const int idx  = (lane < 16) ? (base + 4 * lane)
                                   : (base + 4 * (lane - 16) + 2);
      v2f a;
      a.x = parts[idx];
      a.y = parts[idx + 1];
      // D = A x ones + C : D[m,*] accumulates sum over the 4 partials of row m
      c = __builtin_amdgcn_wmma_f32_16x16x4_f32(
          /*neg_a=*/false, a, /*neg_b=*/false, bones,
          /*c_mod=*/(short)0, c, /*reuse_a=*/false, /*reuse_b=*/false);
    }
    // C/D layout: lane n (0-15) vgpr v = D[M=v, N=n]; lanes 16-31 hold M=8..15.
    // Columns are identical, so lane 0 holds row-sums M=0..7, lane 16 M=8..15.
    float t = c[0] + c[1] + c[2] + c[3] + c[4] + c[5] + c[6] + c[7];
    float thi = __shfl(t, 16, 32);
    if (lane == 0) {
      out[row] = (t + thi) + (float)T * LOGP_CONST;
    }
  }
}

extern "C" void kernel_launch(void* const* d_in, const int* in_sizes, int n_in,
                              void* d_out, int out_size, void* d_ws, size_t ws_size,
                              hipStream_t stream) {
  const float* s = (const float*)d_in[0];
  const float* x = (const float*)d_in[1];
  float* out = (float*)d_out;
  const int B = out_size;                 // 4096 rows
  const int T = in_sizes[0] / B;          // 8192 timesteps
  logistic_logp_kernel<<<B, 256, 0, stream>>>(s, x, out, T);
}